// LlamaAttention_23536420782114
// MI455X (gfx1250) — compile-verified
//
#include <hip/hip_runtime.h>
#include <math.h>

typedef __attribute__((ext_vector_type(2))) float v2f;
typedef __attribute__((ext_vector_type(8))) float v8f;

#define BS      2
#define HEADS   32
#define QLEN    1024
#define KLEN    1024
#define GS      8
#define NG      (HEADS / GS)          // 4
#define NH      (BS * HEADS)          // 64 (bs*head)
#define HEAVY   102                   // int(0.1*1024)
#define RECENT  102
#define NEG_MAXF (-3.4028234663852886e38f)
// density scale = 1 / (bs*head) / (q*(q+1)/2) = 1/64/524800
#define DENS_SCALE (1.0f / 33587200.0f)

// ---------------------------------------------------------------------------
// Kernel 0: zero the group mask and the density accumulator (graph-replay safe)
// ---------------------------------------------------------------------------
__global__ void k_init(unsigned int* __restrict__ gm, float* __restrict__ dens) {
    int i = blockIdx.x * blockDim.x + threadIdx.x;
    if (i < BS * NG * KLEN) gm[i] = 0u;
    if (i == 0) *dens = 0.0f;
}

// ---------------------------------------------------------------------------
// Kernel 1: per-row softmax stats. One wave32 per (bs,h,q) row.
// Each lane holds 32 row elements in registers (8 x float4), wave-reduces
// max and sum(exp) with __shfl_xor (warpSize==32 on gfx1250).
// stats[row] = {rowmax, 1/sumexp}
// ---------------------------------------------------------------------------
__global__ void k_stats(const float* __restrict__ attn, float2* __restrict__ stats) {
    int gw   = (blockIdx.x * blockDim.x + threadIdx.x) >> 5;  // global wave id = row id
    int lane = threadIdx.x & 31;
    if (gw >= NH * QLEN) return;

    const float4* rowp = (const float4*)(attn + (size_t)gw * KLEN);
    float4 v[8];
#pragma unroll
    for (int t = 0; t < 8; ++t) v[t] = rowp[lane + 32 * t];   // coalesced 512B/step

    float mx = v[0].x;
#pragma unroll
    for (int t = 0; t < 8; ++t) {
        mx = fmaxf(mx, fmaxf(fmaxf(v[t].x, v[t].y), fmaxf(v[t].z, v[t].w)));
    }
#pragma unroll
    for (int off = 16; off > 0; off >>= 1) mx = fmaxf(mx, __shfl_xor(mx, off, 32));

    float s = 0.0f;
#pragma unroll
    for (int t = 0; t < 8; ++t) {
        s += __expf(v[t].x - mx) + __expf(v[t].y - mx) +
             __expf(v[t].z - mx) + __expf(v[t].w - mx);
    }
#pragma unroll
    for (int off = 16; off > 0; off >>= 1) s += __shfl_xor(s, off, 32);

    if (lane == 0) stats[gw] = make_float2(mx, 1.0f / s);
}

// ---------------------------------------------------------------------------
// Kernel 2: importance[head,k] = sum_q softmax(attn)[head,q,k] via WMMA.
// D(16x16) += A(16x4 = ones) x B(4x16 = prob tile); after the q loop,
// D[0,n] (lanes 0..15, VGPR0) holds the column sums = importance.
// One wave per (head, 16-wide k tile); exclusive ownership -> plain stores.
// B layout (f32, 4x16): lanes 0-15 hold K=0,1 (VGPR0,1); lanes 16-31 K=2,3.
// ---------------------------------------------------------------------------
__global__ void k_importance(const float* __restrict__ attn,
                             const float2* __restrict__ stats,
                             float* __restrict__ importance) {
    int gw    = (blockIdx.x * blockDim.x + threadIdx.x) >> 5;
    int lane  = threadIdx.x & 31;
    int head  = gw >> 6;            // KLEN/16 = 64 k-tiles per head
    int ktile = gw & 63;

    const float*  base = attn  + (size_t)head * QLEN * KLEN;
    const float2* st   = stats + head * QLEN;

    int col = (ktile << 4) + (lane & 15);
    int rb  = (lane >> 4) << 1;     // lanes 0-15 -> rows q0+0/1, lanes 16-31 -> q0+2/3

    v8f c = {};                     // zero accumulator
    v2f a; a.x = 1.0f; a.y = 1.0f;  // ones A-matrix

    for (int q0 = 0; q0 < QLEN; q0 += 4) {
        int r0 = q0 + rb;
        int r1 = r0 + 1;
        float  x0 = base[(size_t)r0 * KLEN + col];
        float  x1 = base[(size_t)r1 * KLEN + col];
        float2 s0 = st[r0];
        float2 s1 = st[r1];
        v2f b;
        b.x = __expf(x0 - s0.x) * s0.y;
        b.y = __expf(x1 - s1.x) * s1.y;
        // 8 args: (neg_a, A, neg_b, B, c_mod, C, reuse_a, reuse_b)
        c = __builtin_amdgcn_wmma_f32_16x16x4_f32(
                false, a, false, b, (short)0, c, false, false);
    }

    if (lane < 16) importance[head * KLEN + (ktile << 4) + lane] = c[0];
}

// ---------------------------------------------------------------------------
// Kernel 3: per-head top-HEAVY threshold via binary search on value, then
// OR the heavy bits into the per-group mask. One block per (bs,h).
// ---------------------------------------------------------------------------
__global__ void k_topk(const float* __restrict__ importance, unsigned int* __restrict__ gm) {
    __shared__ float simp[KLEN];
    __shared__ int   scount;
    int headlin = blockIdx.x;       // 0..NH-1
    int tid     = threadIdx.x;      // 256 threads

    const float* row = importance + headlin * KLEN;
    for (int i = tid; i < KLEN; i += 256) simp[i] = row[i];
    __syncthreads();

    // importance values lie in [0, QLEN]; binary-search the 102nd-largest
    float lo = 0.0f, hi = (float)(QLEN + 1);
    for (int it = 0; it < 30; ++it) {
        if (tid == 0) scount = 0;
        __syncthreads();
        float mid = 0.5f * (lo + hi);
        int cnt = 0;
        for (int i = tid; i < KLEN; i += 256) cnt += (simp[i] >= mid) ? 1 : 0;
        atomicAdd(&scount, cnt);
        __syncthreads();
        if (scount >= HEAVY) lo = mid; else hi = mid;
        __syncthreads();
    }

    int bs = headlin >> 5;          // / HEADS
    int h  = headlin & 31;
    int ng = h >> 3;                // / GS
    unsigned int* g = gm + (bs * NG + ng) * KLEN;
    for (int i = tid; i < KLEN; i += 256)
        if (simp[i] >= lo) atomicOr(&g[i], 1u);
}

// ---------------------------------------------------------------------------
// Kernel 4: write fetch_mask and accumulate density.
// One block per (bs,h,q) row; each thread emits one float4 of k.
// keep = (gm[k] & causal) | (causal & k >= q-RECENT)
// ---------------------------------------------------------------------------
__global__ void k_mask(const unsigned int* __restrict__ gm,
                       float* __restrict__ out, float* __restrict__ dens) {
    int rowid   = blockIdx.x;             // 0 .. NH*QLEN-1
    int q       = rowid & (QLEN - 1);
    int headlin = rowid >> 10;
    int bs      = headlin >> 5;
    int h       = headlin & 31;
    int ng      = h >> 3;
    const unsigned int* g = gm + (bs * NG + ng) * KLEN;

    float4* orow = (float4*)(out + (size_t)rowid * KLEN);
    int tid   = threadIdx.x;              // 256 threads
    int kbase = tid << 2;
    int rlo   = q - RECENT;

    int cnt = 0;
    float4 v;
#pragma unroll
    for (int j = 0; j < 4; ++j) {
        int  k      = kbase + j;
        bool causal = (k <= q);
        bool keep   = causal && ((g[k] != 0u) || (k >= rlo));
        cnt += keep ? 1 : 0;
        ((float*)&v)[j] = keep ? 0.0f : NEG_MAXF;
    }
    orow[tid] = v;

    // reduce keep-count: wave32 shuffle reduce, then one LDS atomic per wave
    __shared__ int sc;
    if (tid == 0) sc = 0;
    __syncthreads();
#pragma unroll
    for (int off = 16; off > 0; off >>= 1) cnt += __shfl_xor(cnt, off, 32);
    if ((tid & 31) == 0) atomicAdd(&sc, cnt);
    __syncthreads();
    if (tid == 0) atomicAdd(dens, (float)sc * DENS_SCALE);
}

// ---------------------------------------------------------------------------
extern "C" void kernel_launch(void* const* d_in, const int* in_sizes, int n_in,
                              void* d_out, int out_size, void* d_ws, size_t ws_size,
                              hipStream_t stream) {
    const float* attn = (const float*)d_in[0];
    float* out = (float*)d_out;

    // workspace layout
    char* ws = (char*)d_ws;
    float*        importance = (float*)ws;                                   // 256 KB
    float2*       stats      = (float2*)(ws + (size_t)NH * KLEN * 4);        // 512 KB
    unsigned int* gm         = (unsigned int*)(ws + (size_t)NH * KLEN * 4
                                                  + (size_t)NH * QLEN * 8);  // 32 KB
    float* dens = out + (size_t)out_size - 1;   // density scalar after the mask

    k_init<<<(BS * NG * KLEN + 255) / 256, 256, 0, stream>>>(gm, dens);
    // one wave per row, 8 waves per block
    k_stats<<<(NH * QLEN) / 8, 256, 0, stream>>>(attn, stats);
    // one wave per (head, ktile): NH*64 waves, 4 waves per block
    k_importance<<<(NH * 64) / 4, 128, 0, stream>>>(attn, stats, importance);
    k_topk<<<NH, 256, 0, stream>>>(importance, gm);
    k_mask<<<NH * QLEN, 256, 0, stream>>>(gm, out, dens);
}